// QuantAct_14525579395606
// MI455X (gfx1250) — compile-verified
//
#include <hip/hip_runtime.h>
#include <math.h>

// ---------------------------------------------------------------------------
// QuantAct requantize-add-quantize-dequantize, MI455X (gfx1250).
// Elementwise, HBM-bound: 402 MB traffic -> ~17us floor at 23.3 TB/s.
// Strategy: deep async global->LDS pipeline (ASYNCcnt): each wave keeps 8KB
// (16 x async b128) of loads in flight without VGPR pressure, consumes stages
// via ds_load_b128, computes the exact int64 multiply-shift-round with
// wave-uniform shift parameters, and writes with nontemporal b128 stores.
// ---------------------------------------------------------------------------

#define ACC 23
#define WAVES_PER_BLOCK 8
#define THREADS (32 * WAVES_PER_BLOCK)
#define STAGES 8
#define MAX_BLOCKS 4096

typedef float f32x4 __attribute__((ext_vector_type(4)));
typedef int v4i __attribute__((ext_vector_type(4)));

#define __AS1 __attribute__((address_space(1)))
#define __AS3 __attribute__((address_space(3)))

// Detect the gfx1250 async global->LDS builtins (device pass only; on the
// host pass __has_builtin() is 0 and the fallback body is parsed instead).
#if defined(__has_builtin)
#if __has_builtin(__builtin_amdgcn_global_load_async_to_lds_b128) && \
    __has_builtin(__builtin_amdgcn_s_wait_asynccnt)
#define USE_ASYNC_LDS 1
#else
#define USE_ASYNC_LDS 0
#endif
#else
#define USE_ASYNC_LDS 0
#endif

struct QParams {
  int nm_x;   // 24-bit quantized mantissa of pre_act_scale / out_scale
  int e_x;    // frexp exponent
  int nm_id;
  int e_id;
  float scale;
};

// One-thread setup: float32 scale exactly as the reference computes it, then
// float64 frexp decomposition and round-to-nearest-even mantissa quantization
// (matches jnp.round / torch default rounding).
__global__ void qa_setup(const float* __restrict__ pre_sf,
                         const float* __restrict__ id_sf,
                         const float* __restrict__ xmin,
                         const float* __restrict__ xmax,
                         QParams* __restrict__ qp,
                         float* __restrict__ scale_out) {
  float amax = fmaxf(fabsf(xmin[0]), fabsf(xmax[0]));
  if (amax < 1e-8f) amax = 1e-8f;
  float scale = amax / 127.0f;           // float32, per reference
  double s64 = (double)scale;
  double ns_x = (double)pre_sf[0] / s64;
  double ns_id = (double)id_sf[0] / s64;
  int ex, eid;
  double mx = frexp(ns_x, &ex);
  double mid = frexp(ns_id, &eid);
  qp->nm_x = (int)llrint(mx * (double)(1 << ACC));
  qp->e_x = ex;
  qp->nm_id = (int)llrint(mid * (double)(1 << ACC));
  qp->e_id = eid;
  qp->scale = scale;
  *scale_out = scale;                    // second reference output
}

// Bit-exact integer multiply-shift-round. |xi| <= 2^31, nm < 2^24 -> product
// fits int64. Shift direction/amount are wave-uniform (scalar branch).
__device__ __forceinline__ long long qa_fpmul(int xi, int nm, int e) {
  long long t = (long long)xi * (long long)nm;
  int el = e - ACC;
  if (el >= 0) {
    int sl = el > 62 ? 62 : el;
    return t << sl;                      // left-shift branch (e - ACC >= 0)
  }
  int sr = -el;
  if (sr > 62) sr = 62;
  // round-half-away-from-zero: nudge = (1 << (sr-1)) - (t < 0)
  long long nudge = (1LL << (sr - 1)) - (long long)(t < 0);
  return (t + nudge) >> sr;
}

__device__ __forceinline__ float qa_one(float xf, float idf, int nm_x, int e_x,
                                        int nm_id, int e_id, float scale) {
  long long s = qa_fpmul((int)xf, nm_x, e_x) + qa_fpmul((int)idf, nm_id, e_id);
  s = s < -128 ? -128 : (s > 127 ? 127 : s);   // clip to int8 range
  return (float)(int)s * scale;                // dequantize
}

__device__ __forceinline__ f32x4 qa_vec4(f32x4 xr, f32x4 ir, int nm_x, int e_x,
                                         int nm_id, int e_id, float scale) {
  f32x4 o;
  o.x = qa_one(xr.x, ir.x, nm_x, e_x, nm_id, e_id, scale);
  o.y = qa_one(xr.y, ir.y, nm_x, e_x, nm_id, e_id, scale);
  o.z = qa_one(xr.z, ir.z, nm_x, e_x, nm_id, e_id, scale);
  o.w = qa_one(xr.w, ir.w, nm_x, e_x, nm_id, e_id, scale);
  return o;
}

__global__ __launch_bounds__(THREADS) void qa_main(
    const float* __restrict__ x, const float* __restrict__ idn,
    const QParams* __restrict__ qp, float* __restrict__ out, long long nvec,
    long long ntot) {
  const int nm_x = qp->nm_x, e_x = qp->e_x;
  const int nm_id = qp->nm_id, e_id = qp->e_id;
  const float scale = qp->scale;

  const f32x4* __restrict__ xv = (const f32x4*)x;
  const f32x4* __restrict__ iv = (const f32x4*)idn;
  f32x4* __restrict__ ov = (f32x4*)out;

#if USE_ASYNC_LDS
  // Per-wave 8-stage pipeline, 1KB/stage (x + identity), 64KB LDS per block.
  __shared__ f32x4 stg[WAVES_PER_BLOCK][STAGES][2][32];
  const int lane = (int)(threadIdx.x & 31u);
  const int wave = (int)(threadIdx.x >> 5);
  const long long wstride = 32LL * (long long)gridDim.x * WAVES_PER_BLOCK;
  const long long base0 = 32LL * ((long long)blockIdx.x * WAVES_PER_BLOCK + wave);

  // Prologue: fill all stages (up to 16 async loads = 8KB in flight / wave).
#pragma unroll
  for (int s = 0; s < STAGES; ++s) {
    long long idx = base0 + (long long)s * wstride + lane;
    if (idx < nvec) {
      __builtin_amdgcn_global_load_async_to_lds_b128(
          (__AS1 v4i*)(xv + idx), (__AS3 v4i*)&stg[wave][s][0][lane], 0, 0);
      __builtin_amdgcn_global_load_async_to_lds_b128(
          (__AS1 v4i*)(iv + idx), (__AS3 v4i*)&stg[wave][s][1][lane], 0, 0);
    }
  }

  int stage = 0;
  for (long long b = base0; b < nvec; b += wstride) {
    // Allow the 2*(STAGES-1) newer loads to stay in flight; the two loads of
    // the current stage (oldest, loads complete in order) are done after this.
    __builtin_amdgcn_s_wait_asynccnt(2 * (STAGES - 1));
    asm volatile("" ::: "memory");

    long long idx = b + lane;
    if (idx < nvec) {
      f32x4 xr = stg[wave][stage][0][lane];   // ds_load_b128
      f32x4 ir = stg[wave][stage][1][lane];
      f32x4 o = qa_vec4(xr, ir, nm_x, e_x, nm_id, e_id, scale);
      __builtin_nontemporal_store(o, ov + idx);  // streaming store
    }

    // Make sure the DS reads of this slot completed before async refill
    // overwrites it (DS pipe vs async DMA writes are not ordered).
    asm volatile("s_wait_dscnt 0x0" ::: "memory");

    long long nidx = b + (long long)STAGES * wstride + lane;
    if (nidx < nvec) {
      __builtin_amdgcn_global_load_async_to_lds_b128(
          (__AS1 v4i*)(xv + nidx), (__AS3 v4i*)&stg[wave][stage][0][lane], 0, 0);
      __builtin_amdgcn_global_load_async_to_lds_b128(
          (__AS1 v4i*)(iv + nidx), (__AS3 v4i*)&stg[wave][stage][1][lane], 0, 0);
    }
    stage = (stage + 1) & (STAGES - 1);
  }
#else
  // Fallback: vectorized b128 grid-stride with far-ahead prefetch
  // (global_prefetch_b8) and nontemporal streaming loads/stores.
  const long long stride = (long long)gridDim.x * THREADS;
  for (long long i = (long long)blockIdx.x * THREADS + threadIdx.x; i < nvec;
       i += stride) {
    long long pf = i + 8 * stride;
    if (pf < nvec) {
      __builtin_prefetch((const void*)(xv + pf), 0, 1);
      __builtin_prefetch((const void*)(iv + pf), 0, 1);
    }
    f32x4 xr = __builtin_nontemporal_load(xv + i);
    f32x4 ir = __builtin_nontemporal_load(iv + i);
    f32x4 o = qa_vec4(xr, ir, nm_x, e_x, nm_id, e_id, scale);
    __builtin_nontemporal_store(o, ov + i);
  }
#endif

  // Scalar tail for element counts not divisible by 4 (not hit for the
  // reference shape 4*4096*2048, but keeps the kernel general).
  if (blockIdx.x == 0) {
    for (long long i = nvec * 4 + threadIdx.x; i < ntot; i += THREADS) {
      out[i] = qa_one(x[i], idn[i], nm_x, e_x, nm_id, e_id, scale);
    }
  }
}

extern "C" void kernel_launch(void* const* d_in, const int* in_sizes, int n_in,
                              void* d_out, int out_size, void* d_ws,
                              size_t ws_size, hipStream_t stream) {
  // setup_inputs() order: x, identity, pre_act_scaling_factor,
  // identity_scaling_factor, x_min, x_max
  const float* x = (const float*)d_in[0];
  const float* idn = (const float*)d_in[1];
  const float* pre_sf = (const float*)d_in[2];
  const float* id_sf = (const float*)d_in[3];
  const float* xmin = (const float*)d_in[4];
  const float* xmax = (const float*)d_in[5];

  long long ntot = (long long)in_sizes[0];
  float* out = (float*)d_out;
  QParams* qp = (QParams*)d_ws;
  float* scale_out = out + (out_size - 1);  // outputs: [dequant(N), scale(1)]

  qa_setup<<<1, 1, 0, stream>>>(pre_sf, id_sf, xmin, xmax, qp, scale_out);

  long long nvec = ntot / 4;
  long long want = (nvec + (32LL * WAVES_PER_BLOCK) - 1) / (32LL * WAVES_PER_BLOCK);
  int blocks = (int)(want > MAX_BLOCKS ? MAX_BLOCKS : (want < 1 ? 1 : want));

  qa_main<<<blocks, THREADS, 0, stream>>>(x, idn, qp, out, nvec, ntot);
}